// CondNMSPostProcess_13408887899033
// MI455X (gfx1250) — compile-verified
//
#include <hip/hip_runtime.h>

#define NQ      900
#define TOPK    100
#define KEEPN   20
#define NPAD    112     // 7 * 16 tiles
#define SORTN   1024
#define IOU_T   0.7f

typedef float v2f __attribute__((ext_vector_type(2)));
typedef float v8f __attribute__((ext_vector_type(8)));

__global__ __launch_bounds__(256) void nms_patch_kernel(
    const float* __restrict__ logits,   // (B,1,N,2)
    const float* __restrict__ boxes,    // (B,1,N,4) cxcywh
    const float* __restrict__ tsizes,   // (B,2)  [h, w]
    float* __restrict__ out,
    int P, int boxOff, int valOff)
{
  __shared__ unsigned long long key[SORTN];                 // packed (score_bits<<32)|(~idx)
  __shared__ float bx0[NPAD], by0[NPAD], bx1[NPAD], by1[NPAD], barea[NPAD];
  __shared__ float vals[TOPK];
  __shared__ unsigned int rmask[NPAD][4];                   // per-row "iou>thresh" bitmask
  __shared__ unsigned int kw[4];

  const int tid = threadIdx.x;
  const int bp  = blockIdx.x;          // batch*P + patch
  const int batch = bp / P;
  const long long qbase = (long long)bp * NQ;

  // CDNA5: prefetch this patch's boxes toward L2 while we sort scores
  for (int q = tid; q < NQ; q += 256)
    __builtin_prefetch(boxes + (qbase + q) * 4, 0, 1);

  // ---- sigmoid scores, pack sortable keys (desc score, tie -> low index) ----
  const float2* lg2 = (const float2*)logits;                // one b64 load per query
  for (int i = tid; i < SORTN; i += 256) {
    unsigned long long k = 0ull;
    if (i < NQ) {
      float x = lg2[qbase + i].y;                           // channel C-1
      float s = 1.0f / (1.0f + __expf(-x));                 // s in (0,1): bits order = value order
      unsigned int sb = __float_as_uint(s);
      k = ((unsigned long long)sb << 32) | (unsigned long long)(0xFFFFFFFFu - (unsigned)i);
    }
    key[i] = k;
  }
  for (int i = tid; i < NPAD * 4; i += 256)
    ((unsigned int*)rmask)[i] = 0u;
  __syncthreads();

  // ---- bitonic sort, descending, 1024 elems / 256 threads ----
  for (int kk2 = 2; kk2 <= SORTN; kk2 <<= 1) {
    for (int j = kk2 >> 1; j > 0; j >>= 1) {
      for (int m = 0; m < SORTN / 256; ++m) {
        int i = tid + (m << 8);
        int l = i ^ j;
        if (l > i) {
          unsigned long long a = key[i], b = key[l];
          bool desc = ((i & kk2) == 0);
          bool sw = desc ? (a < b) : (a > b);
          if (sw) { key[i] = b; key[l] = a; }
        }
      }
      __syncthreads();
    }
  }

  // ---- gather top-100 boxes, cxcywh -> xyxy, scale; pad rows inert ----
  float img_h = tsizes[batch * 2 + 0];
  float img_w = tsizes[batch * 2 + 1];
  if (tid < NPAD) {
    float x0, y0, x1, y1;
    if (tid < TOPK) {
      unsigned long long kk = key[tid];
      vals[tid] = __uint_as_float((unsigned int)(kk >> 32));
      int q = (int)(0xFFFFFFFFu - (unsigned int)(kk & 0xFFFFFFFFull));
      const float4 b4 = ((const float4*)boxes)[qbase + q];  // global_load_b128
      float cx = b4.x, cy = b4.y, w = b4.z, h = b4.w;
      x0 = (cx - 0.5f * w) * img_w;
      y0 = (cy - 0.5f * h) * img_h;
      x1 = (cx + 0.5f * w) * img_w;
      y1 = (cy + 0.5f * h) * img_h;
    } else {
      x0 = y0 = x1 = y1 = -3.0e30f;   // pad: zero area, zero intersection with anything
    }
    bx0[tid] = x0; by0[tid] = y0; bx1[tid] = x1; by1[tid] = y1;
    barea[tid] = (tid < TOPK) ? (x1 - x0) * (y1 - y0) : 0.0f;
  }
  __syncthreads();

  // ---- IoU in 16x16 tiles: inter on VALU into C-layout, union via WMMA f32 16x16x4 ----
  // union_ij = area_i*1 + 1*area_j + (-inter_ij)  ==>  D = A(16x4) x B(4x16) + C
  const int wave = tid >> 5;
  const int lane = tid & 31;
  for (int tile = wave; tile < 49; tile += 8) {
    int tr = tile / 7, tc = tile % 7;
    int rowBase = tr * 16, colBase = tc * 16;
    int col = colBase + (lane & 15);
    int rh  = (lane >> 4) << 3;                      // 0 or 8: C/D row half per lane group
    float cxa = bx0[col], cya = by0[col], cxb = bx1[col], cyb = by1[col];

    v8f c;
    float inter[8];
    for (int v = 0; v < 8; ++v) {
      int row = rowBase + rh + v;
      float lx = fmaxf(bx0[row], cxa);
      float ly = fmaxf(by0[row], cya);
      float rx = fminf(bx1[row], cxb);
      float ry = fminf(by1[row], cyb);
      float w = fmaxf(rx - lx, 0.0f);
      float h = fmaxf(ry - ly, 0.0f);
      inter[v] = w * h;
      c[v] = -inter[v];
    }
    // A 16x4: lanes 0-15 hold K=0,1 (area_row, 1); lanes 16-31 hold K=2,3 (zeros)
    v2f a, b;
    a[0] = (lane < 16) ? barea[rowBase + lane] : 0.0f;
    a[1] = (lane < 16) ? 1.0f : 0.0f;
    // B 4x16: VGPR0 -> rows K=0 (ones) / K=2 (0); VGPR1 -> rows K=1 (area_col) / K=3 (0)
    b[0] = (lane < 16) ? 1.0f : 0.0f;
    b[1] = (lane < 16) ? barea[colBase + lane] : 0.0f;

    v8f uni = __builtin_amdgcn_wmma_f32_16x16x4_f32(
        false, a, false, b, (short)0, c, false, false);

    unsigned int mball[8];
    for (int v = 0; v < 8; ++v) {
      // iou > T  <=>  inter > T * max(union, 1e-9)   (denominator > 0)
      // avoids the full IEEE v_div_scale/v_div_fixup sequence per element
      bool hit = inter[v] > IOU_T * fmaxf(uni[v], 1e-9f);
      mball[v] = (unsigned int)__ballot(hit);
    }
    if (lane == 0) {
      int word  = colBase >> 5;
      int shift = colBase & 31;                       // 0 or 16
      for (int v = 0; v < 8; ++v) {
        atomicOr(&rmask[rowBase + v][word],     (mball[v] & 0xFFFFu) << shift);
        atomicOr(&rmask[rowBase + 8 + v][word], (mball[v] >> 16)     << shift);
      }
    }
  }
  __syncthreads();

  // ---- greedy NMS over bitmasks (inherently serial, 100 iters on 4 words) ----
  if (tid == 0) {
    unsigned int kk[4] = {0xFFFFFFFFu, 0xFFFFFFFFu, 0xFFFFFFFFu, 0x0000000Fu}; // 100 bits
    for (int i = 0; i < TOPK; ++i) {
      int wi = i >> 5, bit = i & 31;
      if ((kk[wi] >> bit) & 1u) {
        for (int w = 0; w < 4; ++w) {
          unsigned int sup = rmask[i][w];
          if (w < wi)       sup = 0u;
          else if (w == wi) sup &= (bit == 31) ? 0u : (0xFFFFFFFFu << (bit + 1)); // only j > i
          kk[w] &= ~sup;
        }
      }
    }
    kw[0] = kk[0]; kw[1] = kk[1]; kw[2] = kk[2]; kw[3] = kk[3];
  }
  __syncthreads();

  // ---- stable reorder (kept first, original order) and emit first 20 ----
  if (tid < TOPK) {
    unsigned int kv[4] = {kw[0], kw[1], kw[2], kw[3]};
    int total = __popc(kv[0]) + __popc(kv[1]) + __popc(kv[2]) + __popc(kv[3]);
    int wi = tid >> 5, bit = tid & 31;
    int before = 0;
    for (int w = 0; w < wi; ++w) before += __popc(kv[w]);
    before += __popc(kv[wi] & ((1u << bit) - 1u));
    bool kept = (kv[wi] >> bit) & 1u;
    int rank = kept ? before : (total + (tid - before));
    if (rank < KEEPN) {
      int slot = bp * KEEPN + rank;
      out[slot] = vals[tid];
      float* ob = out + boxOff + (long long)slot * 4;
      ob[0] = bx0[tid]; ob[1] = by0[tid]; ob[2] = bx1[tid]; ob[3] = by1[tid];
      out[valOff + slot] = kept ? 1.0f : 0.0f;
    }
  }
}

extern "C" void kernel_launch(void* const* d_in, const int* in_sizes, int n_in,
                              void* d_out, int out_size, void* d_ws, size_t ws_size,
                              hipStream_t stream) {
  const float* logits = (const float*)d_in[0];
  const float* boxes  = (const float*)d_in[1];
  const float* tsz    = (const float*)d_in[2];
  float* out = (float*)d_out;

  int B = in_sizes[2] / 2;                                  // target_sizes (B,2)
  long long N = (long long)in_sizes[0] / (2LL * B);         // logits (B,1,N,2)
  int P = (int)(N / NQ);
  int nPatch = B * P;
  int scN    = nPatch * KEEPN;      // scores element count
  int boxOff = scN;                 // boxes start
  int valOff = scN + scN * 4;       // valid start

  nms_patch_kernel<<<dim3(nPatch), dim3(256), 0, stream>>>(
      logits, boxes, tsz, out, P, boxOff, valOff);
}